// Attention_39204461478201
// MI455X (gfx1250) — compile-verified
//
#include <hip/hip_runtime.h>

typedef __attribute__((ext_vector_type(2))) float v2f;
typedef __attribute__((ext_vector_type(8))) float v8f;

#define BATCH  16
#define TSEQ   2048
#define HIDDEN 512
#define KSEG   8
#define KPER   (TSEQ / KSEG)   // 256 time steps per wave

// ---------------------------------------------------------------------------
// out[b,h] = mean_k X[b,k,h]  (exact algebraic reduction of the reference:
// softmax over axis=1 followed by mean over axis=1 makes the attention weights
// integrate to 1 per key, so context-mean == time-mean of X).
//
// Implemented as D = ones(16x4) · X_tile(4x16) accumulated on the f32 WMMA
// pipe; every row of D carries the 16 column sums for one hidden tile.
// ---------------------------------------------------------------------------

__global__ void zero_out_kernel(float* __restrict__ out, int n) {
    int i = blockIdx.x * blockDim.x + threadIdx.x;
    if (i < n) out[i] = 0.0f;
}

__global__ __launch_bounds__(32)
void mean_time_wmma_kernel(const float* __restrict__ x, float* __restrict__ out) {
    const int lane = threadIdx.x;                 // 0..31, one wave per block

    int idx = blockIdx.x;
    const int kseg = idx & (KSEG - 1); idx >>= 3; // k segment     0..7
    const int ht   = idx & 31;         idx >>= 5; // hidden tile   0..31
    const int b    = idx;                         // batch         0..15

    // B-matrix (4x16 f32) lane mapping: n = lane%16 (column / hidden index),
    // lanes 0-15 hold K={0,1} in {V0,V1}, lanes 16-31 hold K={2,3}.
    const int n  = lane & 15;
    const int kk = (lane >> 4) * 2;

    // A = all-ones 16x4: every lane holds 1.0 in both A registers.
    v2f a; a[0] = 1.0f; a[1] = 1.0f;

    v8f acc0 = {}, acc1 = {}, acc2 = {}, acc3 = {};

    const float* p = x + (size_t)b * TSEQ * HIDDEN
                       + (size_t)(kseg * KPER + kk) * HIDDEN
                       + ht * 16 + n;

    // 4 independent accumulator chains -> WMMA RAW hazard never serializes,
    // 8 global_load_b32 in flight per unrolled iteration.
    #pragma unroll 1
    for (int k = 0; k < KPER; k += 16) {
        v2f b0, b1, b2, b3;
        b0[0] = p[0 * HIDDEN];  b0[1] = p[1 * HIDDEN];
        b1[0] = p[4 * HIDDEN];  b1[1] = p[5 * HIDDEN];
        b2[0] = p[8 * HIDDEN];  b2[1] = p[9 * HIDDEN];
        b3[0] = p[12 * HIDDEN]; b3[1] = p[13 * HIDDEN];

        acc0 = __builtin_amdgcn_wmma_f32_16x16x4_f32(false, a, false, b0,
                                                     (short)0, acc0, false, false);
        acc1 = __builtin_amdgcn_wmma_f32_16x16x4_f32(false, a, false, b1,
                                                     (short)0, acc1, false, false);
        acc2 = __builtin_amdgcn_wmma_f32_16x16x4_f32(false, a, false, b2,
                                                     (short)0, acc2, false, false);
        acc3 = __builtin_amdgcn_wmma_f32_16x16x4_f32(false, a, false, b3,
                                                     (short)0, acc3, false, false);

        p += 16 * HIDDEN;
    }

    // D row 0: lane n (n<16), VGPR0 holds sum over this wave's K range for
    // hidden column ht*16+n (all D rows are identical since A == ones).
    const float partial = (acc0[0] + acc1[0]) + (acc2[0] + acc3[0]);

    if (lane < 16) {
        atomicAdd(out + b * HIDDEN + ht * 16 + n, partial * (1.0f / (float)TSEQ));
    }
}

extern "C" void kernel_launch(void* const* d_in, const int* in_sizes, int n_in,
                              void* d_out, int out_size, void* d_ws, size_t ws_size,
                              hipStream_t stream) {
    const float* x  = (const float*)d_in[0];     // [16, 2048, 512] f32
    float* out      = (float*)d_out;             // [16, 512] f32

    // d_out is poisoned by the harness and atomics accumulate: zero it first
    // (stream order guarantees this completes before the reduction kernel).
    zero_out_kernel<<<(out_size + 255) / 256, 256, 0, stream>>>(out, out_size);

    const int blocks = BATCH * (HIDDEN / 16) * KSEG;   // 4096 single-wave WGs
    mean_time_wmma_kernel<<<blocks, 32, 0, stream>>>(x, out);
}